// WeightedNeighborSampler_42339787604731
// MI455X (gfx1250) — compile-verified
//
#include <hip/hip_runtime.h>
#include <hip/hip_bf16.h>
#include <stdint.h>

// WeightedNeighborSampler for MI455X (gfx1250, wave32).
//
// Design (memory-bound; ~45 MB total traffic -> ~2us at 23.3 TB/s):
//  - 1 wave32 per batch row; 8 waves (256 threads) per block.
//  - CDNA5 async DMA: global_load_async_to_lds_b128 stages the 512B weight
//    row into LDS (ASYNCcnt / s_wait_asynccnt), no VGPR round trip.
//  - Each lane owns 4 contiguous weights -> lane-local prefix, then a
//    5-step __shfl_up wave32 scan builds the 128-entry inclusive CDF in LDS.
//  - Lanes 0..num_samples-1 draw a deterministic counter-hash uniform and
//    binary-search the CDF (7 steps), then gather the neighbor id.
//  - global_prefetch (builtin_prefetch) warms the neighbor-id row.

#define MAX_DEGREE      128
#define WAVES_PER_BLOCK 8
#define BLOCK_THREADS   (WAVES_PER_BLOCK * 32)

__device__ __forceinline__ uint32_t wns_hash(uint32_t x) {
  x ^= x >> 16; x *= 0x7feb352du;
  x ^= x >> 15; x *= 0x846ca68bu;
  x ^= x >> 16;
  return x;
}

__global__ __launch_bounds__(BLOCK_THREADS) void
WeightedNeighborSampler_42339787604731_kernel(
    const int*   __restrict__ ids,
    const int*   __restrict__ adj_info,
    const float* __restrict__ adj_weight,
    const int*   __restrict__ num_samples_p,
    int*         __restrict__ out,
    int batch)
{
  __shared__ __align__(16) float s_cum[WAVES_PER_BLOCK][MAX_DEGREE];

  const int lane = (int)(threadIdx.x & 31u);
  const int wave = (int)(threadIdx.x >> 5u);
  const int b    = blockIdx.x * WAVES_PER_BLOCK + wave;
  if (b >= batch) return;

  const int nsamp = num_samples_p[0];
  const int row   = ids[b];
  const float* wrow = adj_weight + (size_t)row * MAX_DEGREE;
  const int*   nrow = adj_info   + (size_t)row * MAX_DEGREE;

  // Warm the neighbor-id row (gathered at the end): lowers to global_prefetch_b8.
  __builtin_prefetch(nrow + lane * 4, 0, 1);

  // ---- CDNA5 async copy: stage the 512B weight row, global -> LDS ----
  // Per-lane: 16B from wrow[lane*4 .. lane*4+3] into this wave's LDS slot.
  {
    uint32_t lds_dst = (uint32_t)(uintptr_t)&s_cum[wave][lane * 4];
    uint64_t gaddr   = (uint64_t)(uintptr_t)(wrow + lane * 4);
    asm volatile("global_load_async_to_lds_b128 %0, %1, off"
                 :: "v"(lds_dst), "v"(gaddr) : "memory");
    asm volatile("s_wait_asynccnt 0x0" ::: "memory");
  }
  __builtin_amdgcn_wave_barrier();

  // Each lane picks up its 4 staged weights.
  float4 w = *reinterpret_cast<const float4*>(&s_cum[wave][lane * 4]);

  // Lane-local inclusive prefix over 4 elements.
  float c0 = w.x;
  float c1 = c0 + w.y;
  float c2 = c1 + w.z;
  float c3 = c2 + w.w;

  // wave32 inclusive scan of lane totals (5 shuffle steps).
  float incl = c3;
  #pragma unroll
  for (int d = 1; d < 32; d <<= 1) {
    float t = __shfl_up(incl, d, 32);
    if (lane >= d) incl += t;
  }
  const float excl  = incl - c3;
  const float total = __shfl(incl, 31, 32);

  // Publish the 128-entry inclusive CDF to LDS for the search phase.
  float4 cum = make_float4(c0 + excl, c1 + excl, c2 + excl, c3 + excl);
  *reinterpret_cast<float4*>(&s_cum[wave][lane * 4]) = cum;
  asm volatile("s_wait_dscnt 0x0" ::: "memory");
  __builtin_amdgcn_wave_barrier();

  // ---- Sampling: deterministic counter-hash -> inverse-CDF search ----
  const float* cumrow = &s_cum[wave][0];
  const float  inv24  = 1.0f / 16777216.0f;  // 2^-24
  for (int s = lane; s < nsamp; s += 32) {
    uint32_t r = wns_hash(0x9e3779b9u ^ (uint32_t)(b * 4097 + s * 131071));
    float u = (float)(r >> 8) * inv24 * total;   // u in [0, total)

    // lower-bound binary search over 128 inclusive-cumsum entries.
    int idx = 0;
    #pragma unroll
    for (int sh = 6; sh >= 0; --sh) {
      int nxt = idx + (1 << sh);
      if (cumrow[nxt - 1] <= u) idx = nxt;
    }
    if (idx > MAX_DEGREE - 1) idx = MAX_DEGREE - 1;

    out[(size_t)b * nsamp + s] = nrow[idx];
  }
}

extern "C" void kernel_launch(void* const* d_in, const int* in_sizes, int n_in,
                              void* d_out, int out_size, void* d_ws, size_t ws_size,
                              hipStream_t stream) {
  (void)n_in; (void)out_size; (void)d_ws; (void)ws_size;

  const int*   ids         = (const int*)d_in[0];     // [BATCH] int32
  const int*   adj_info    = (const int*)d_in[1];     // [N_NODES, 128] int32
  const float* adj_weight  = (const float*)d_in[2];   // [N_NODES, 128] float32
  const int*   num_samples = (const int*)d_in[3];     // scalar int (25)

  const int batch  = in_sizes[0];
  const int blocks = (batch + WAVES_PER_BLOCK - 1) / WAVES_PER_BLOCK;

  hipLaunchKernelGGL(WeightedNeighborSampler_42339787604731_kernel,
                     dim3(blocks), dim3(BLOCK_THREADS), 0, stream,
                     ids, adj_info, adj_weight, num_samples,
                     (int*)d_out, batch);
}